// GHMC_Loss_29583734734927
// MI455X (gfx1250) — compile-verified
//
#include <hip/hip_runtime.h>

// GHM-C loss, single pass over memory.
//   Pass 1 (ghm_partial): stream pred/gt once (NT b128 loads, 2x-unrolled
//     grid-stride for MLP), per-thread register histograms of
//     (count_b, sum_b bce), LDS tree reduce, global f64/u32 atomics into ws.
//   Pass 2 (ghm_final): 10-bin weighting + mean, writes the scalar.
// Memory roofline: 268 MB @ 23.3 TB/s ~= 11.5 us; compute is ~balanced, so
// single-pass (vs the reference's two passes) is the key win.

#define GHM_BINS 10
#define GHM_EPS 1e-4f
#define GHM_BLOCK 256

typedef float v4f __attribute__((ext_vector_type(4)));

__device__ __forceinline__ void ghm_accum(float x, float t,
                                          float* __restrict__ acc,
                                          unsigned* __restrict__ cnt) {
    // Shared E = exp(-|x|) feeds both sigmoid and softplus.
    float ax = __builtin_fabsf(x);
    float E  = __expf(-ax);                        // v_exp_f32 path
    float r  = __builtin_amdgcn_rcpf(1.0f + E);    // 1/(1+E), ~1 ulp
    float s  = (x >= 0.0f) ? r : (1.0f - r);       // sigmoid(x)
    float g  = __builtin_fabsf(s - t);             // gradient magnitude
    int bin  = (int)(g * (10.0f - GHM_EPS));       // trunc == floor (g >= 0)
    bin = bin < 0 ? 0 : (bin > GHM_BINS - 1 ? GHM_BINS - 1 : bin);
    // bce = logaddexp(0, x) - x*t = log(1+exp(-|x|)) + max(x,0) - x*t
    float bce = __logf(1.0f + E) + fmaxf(x, 0.0f) - x * t;
    #pragma unroll
    for (int b = 0; b < GHM_BINS; ++b) {           // fully unrolled -> VGPRs
        bool h = (bin == b);
        cnt[b] += h ? 1u : 0u;
        acc[b] += h ? bce : 0.0f;
    }
}

__device__ __forceinline__ void ghm_accum4(v4f p, v4f t,
                                           float* __restrict__ acc,
                                           unsigned* __restrict__ cnt) {
    ghm_accum(p.x, t.x, acc, cnt);
    ghm_accum(p.y, t.y, acc, cnt);
    ghm_accum(p.z, t.z, acc, cnt);
    ghm_accum(p.w, t.w, acc, cnt);
}

__global__ void ghm_init(double* __restrict__ g_sum, unsigned* __restrict__ g_cnt) {
    int t = threadIdx.x;
    if (t < GHM_BINS) { g_sum[t] = 0.0; g_cnt[t] = 0u; }
}

__global__ __launch_bounds__(GHM_BLOCK)
void ghm_partial(const float* __restrict__ pred, const float* __restrict__ gt,
                 long long n, double* __restrict__ g_sum, unsigned* __restrict__ g_cnt) {
    float    acc[GHM_BINS];
    unsigned cnt[GHM_BINS];
    #pragma unroll
    for (int b = 0; b < GHM_BINS; ++b) { acc[b] = 0.0f; cnt[b] = 0u; }

    const long long n4     = n >> 2;
    const long long stride = (long long)gridDim.x * blockDim.x;
    const long long tid_g  = (long long)blockIdx.x * blockDim.x + threadIdx.x;

    const v4f* __restrict__ p4 = (const v4f*)pred;
    const v4f* __restrict__ t4 = (const v4f*)gt;

    // 2x-unrolled grid-stride: 4 b128 loads in flight per iteration.
    long long i = tid_g;
    for (; i + stride < n4; i += 2 * stride) {
        v4f pa = __builtin_nontemporal_load(p4 + i);            // th:NT
        v4f ta = __builtin_nontemporal_load(t4 + i);
        v4f pb = __builtin_nontemporal_load(p4 + i + stride);
        v4f tb = __builtin_nontemporal_load(t4 + i + stride);
        ghm_accum4(pa, ta, acc, cnt);
        ghm_accum4(pb, tb, acc, cnt);
    }
    if (i < n4) {
        v4f pa = __builtin_nontemporal_load(p4 + i);
        v4f ta = __builtin_nontemporal_load(t4 + i);
        ghm_accum4(pa, ta, acc, cnt);
    }
    // Scalar tail (n not divisible by 4).
    const long long base = n4 << 2;
    if (base + tid_g < n) ghm_accum(pred[base + tid_g], gt[base + tid_g], acc, cnt);

    // Block tree reduction in LDS (plain ds ops, no atomic serialization).
    __shared__ float    s_sum[GHM_BINS * GHM_BLOCK];
    __shared__ unsigned s_cnt[GHM_BINS * GHM_BLOCK];
    const int tid = threadIdx.x;
    #pragma unroll
    for (int b = 0; b < GHM_BINS; ++b) {
        s_sum[b * GHM_BLOCK + tid] = acc[b];
        s_cnt[b * GHM_BLOCK + tid] = cnt[b];
    }
    __syncthreads();
    for (int s = GHM_BLOCK / 2; s > 0; s >>= 1) {
        if (tid < s) {
            #pragma unroll
            for (int b = 0; b < GHM_BINS; ++b) {
                s_sum[b * GHM_BLOCK + tid] += s_sum[b * GHM_BLOCK + tid + s];
                s_cnt[b * GHM_BLOCK + tid] += s_cnt[b * GHM_BLOCK + tid + s];
            }
        }
        __syncthreads();
    }
    if (tid < GHM_BINS) {
        atomicAdd(&g_sum[tid], (double)s_sum[tid * GHM_BLOCK]);  // global_atomic_add_f64
        atomicAdd(&g_cnt[tid], s_cnt[tid * GHM_BLOCK]);
    }
}

__global__ void ghm_final(const double* __restrict__ g_sum,
                          const unsigned* __restrict__ g_cnt,
                          float* __restrict__ out, long long n) {
    if (threadIdx.x == 0 && blockIdx.x == 0) {
        int ne = 0;
        #pragma unroll
        for (int b = 0; b < GHM_BINS; ++b) ne += (g_cnt[b] > 0u) ? 1 : 0;
        const float nef = (float)ne;
        double total = 0.0;
        #pragma unroll
        for (int b = 0; b < GHM_BINS; ++b) {
            // Match reference f32 semantics: GD = clip(count*nonempty, EPS); w = 1/GD
            float gd = fmaxf((float)g_cnt[b] * nef, GHM_EPS);
            float w  = 1.0f / gd;
            total += g_sum[b] * (double)w;
        }
        out[0] = (float)(total / (double)n);
    }
}

extern "C" void kernel_launch(void* const* d_in, const int* in_sizes, int n_in,
                              void* d_out, int out_size, void* d_ws, size_t ws_size,
                              hipStream_t stream) {
    const float* pred = (const float*)d_in[0];
    const float* gt   = (const float*)d_in[1];
    const long long n = (long long)in_sizes[0];

    double*   g_sum = (double*)d_ws;
    unsigned* g_cnt = (unsigned*)((char*)d_ws + GHM_BINS * sizeof(double));

    ghm_init<<<1, 32, 0, stream>>>(g_sum, g_cnt);

    const long long n4 = n >> 2;
    long long want = (n4 + GHM_BLOCK - 1) / GHM_BLOCK;
    if (want < 1) want = 1;
    const int grid = (int)(want < 2048 ? want : 2048);

    ghm_partial<<<grid, GHM_BLOCK, 0, stream>>>(pred, gt, n, g_sum, g_cnt);
    ghm_final<<<1, 32, 0, stream>>>(g_sum, g_cnt, (float*)d_out, n);
}